// SGFormer_MIP_60610578481380
// MI455X (gfx1250) — compile-verified
//
#include <hip/hip_runtime.h>
#include <hip/hip_bf16.h>

typedef __attribute__((ext_vector_type(2))) float v2f;
typedef __attribute__((ext_vector_type(8))) float v8f;

static constexpr int NC = 100000;
static constexpr int NV = 200000;
static constexpr int NN = 300000;   // NC + NV
static constexpr int NE = 1000000;
static constexpr int H  = 64;
static constexpr float LN_EPS = 1e-5f;

// ---------------------------------------------------------------------------
// Epilogue modes for the WMMA GEMM
// ---------------------------------------------------------------------------
enum { EPI_NONE = 0, EPI_BIAS = 1, EPI_BIAS_RELU = 2, EPI_LN_RELU = 3, EPI_MIX = 4 };

struct GemmArgs {
    const float* A;     // [rows,64] (or first half of concat)
    const float* A2;    // [rows,64] second concat input (K=128) or null
    const float* W;     // [K,64] row-major
    const float* bias;  // [64] or null
    float*       out;   // [rows,64]
    const float* ln_g;  // [64] LN gamma (EPI_LN_RELU)
    const float* ln_b;  // [64] LN beta
    const float* x1;    // [rows,64] mix input (EPI_MIX)
    float*       dup;   // [rows,64] duplicate output (EPI_MIX)
    int          rows;
};

// Y = epi(X @ W + b).  Wave computes 32 rows (two 16-row M-tiles sharing B
// fragments) x 64 cols with v_wmma_f32_16x16x4_f32; W staged through LDS;
// 8 waves / block -> 256 rows per block.
template<int EPI, bool PRERELU, bool CONCAT>
__global__ __launch_bounds__(256) void gemm64(GemmArgs g) {
    constexpr int K = CONCAT ? 128 : 64;
    __shared__ float Ws[K * 64];
    for (int i = threadIdx.x; i < K * 64; i += 256) Ws[i] = g.W[i];
    __syncthreads();

    const int wave = threadIdx.x >> 5;
    const int lane = threadIdx.x & 31;
    const int r0   = (blockIdx.x * 8 + wave) * 32;
    if (r0 >= g.rows) return;              // rows are multiples of 32
    const int hi = lane >> 4;              // 0 or 1
    const int lo = lane & 15;

    const float* Ar[2]  = {g.A + (size_t)(r0 + lo) * 64,
                           g.A + (size_t)(r0 + 16 + lo) * 64};
    const float* Ar2[2] = {CONCAT ? g.A2 + (size_t)(r0 + lo) * 64 : nullptr,
                           CONCAT ? g.A2 + (size_t)(r0 + 16 + lo) * 64 : nullptr};

    v8f acc[2][4] = {{v8f{}, v8f{}, v8f{}, v8f{}},
                     {v8f{}, v8f{}, v8f{}, v8f{}}};

#pragma unroll
    for (int kk = 0; kk < K / 4; ++kk) {
        const int c0 = 4 * kk + 2 * hi;    // K position of this lane half
        v2f av[2];
#pragma unroll
        for (int mt = 0; mt < 2; ++mt) {
            float a0, a1;
            if (CONCAT) {
                const float* src = (c0 < 64) ? Ar[mt] : (Ar2[mt] - 64);
                a0 = src[c0]; a1 = src[c0 + 1];
            } else {
                a0 = Ar[mt][c0]; a1 = Ar[mt][c0 + 1];
            }
            if (PRERELU) { a0 = fmaxf(a0, 0.f); a1 = fmaxf(a1, 0.f); }
            av[mt] = v2f{a0, a1};
        }
#pragma unroll
        for (int t = 0; t < 4; ++t) {
            v2f bv = {Ws[c0 * 64 + t * 16 + lo], Ws[(c0 + 1) * 64 + t * 16 + lo]};
            acc[0][t] = __builtin_amdgcn_wmma_f32_16x16x4_f32(
                false, av[0], false, bv, (short)0, acc[0][t], false, false);
            acc[1][t] = __builtin_amdgcn_wmma_f32_16x16x4_f32(
                false, av[1], false, bv, (short)0, acc[1][t], false, false);
        }
    }

    float bcol[4];
#pragma unroll
    for (int t = 0; t < 4; ++t) bcol[t] = g.bias ? g.bias[t * 16 + lo] : 0.f;

    if (EPI == EPI_LN_RELU) {
        float gc[4], bc[4];
#pragma unroll
        for (int t = 0; t < 4; ++t) { gc[t] = g.ln_g[t * 16 + lo]; bc[t] = g.ln_b[t * 16 + lo]; }
#pragma unroll
        for (int mt = 0; mt < 2; ++mt) {
#pragma unroll
            for (int r = 0; r < 8; ++r) {
                float v[4], sum = 0.f, ss = 0.f;
#pragma unroll
                for (int t = 0; t < 4; ++t) { v[t] = acc[mt][t][r] + bcol[t]; sum += v[t]; ss += v[t] * v[t]; }
                for (int m = 1; m < 16; m <<= 1) { sum += __shfl_xor(sum, m, 32); ss += __shfl_xor(ss, m, 32); }
                const float mean = sum * (1.f / 64.f);
                const float var  = ss * (1.f / 64.f) - mean * mean;
                const float rstd = rsqrtf(var + LN_EPS);
                float* o = g.out + (size_t)(r0 + mt * 16 + r + 8 * hi) * 64;
#pragma unroll
                for (int t = 0; t < 4; ++t)
                    o[t * 16 + lo] = fmaxf((v[t] - mean) * rstd * gc[t] + bc[t], 0.f);
            }
        }
    } else if (EPI == EPI_MIX) {
#pragma unroll
        for (int mt = 0; mt < 2; ++mt) {
#pragma unroll
            for (int r = 0; r < 8; ++r) {
                const size_t row = (size_t)(r0 + mt * 16 + r + 8 * hi) * 64;
#pragma unroll
                for (int t = 0; t < 4; ++t) {
                    const int c = t * 16 + lo;
                    const float vv = fmaxf(acc[mt][t][r] + bcol[t], 0.f);
                    const float y  = 0.8f * vv + 0.2f * g.x1[row + c];
                    g.out[row + c] = y;
                    g.dup[row + c] = y;
                }
            }
        }
    } else {
#pragma unroll
        for (int mt = 0; mt < 2; ++mt) {
#pragma unroll
            for (int r = 0; r < 8; ++r) {
                float* o = g.out + (size_t)(r0 + mt * 16 + r + 8 * hi) * 64;
#pragma unroll
                for (int t = 0; t < 4; ++t) {
                    float vv = acc[mt][t][r] + bcol[t];
                    if (EPI == EPI_BIAS_RELU) vv = fmaxf(vv, 0.f);
                    o[t * 16 + lo] = vv;
                }
            }
        }
    }
}

// ---------------------------------------------------------------------------
// Edge message kernel: 32 edges per wave (two 16-edge tiles sharing B
// fragments), WMMA against W_final, atomic scatter.
// msg = relu(leftL[li] + rightL[ri] + eh*we) @ Wf + bf ; agg[ri] += msg
// ---------------------------------------------------------------------------
struct EdgeArgs {
    const int*   li;     // [E]
    const int*   ri;     // [E]
    const float* eh;     // [E]
    const float* leftL;  // [nleft,64]  (left linear, bias folded)
    const float* rightL; // [nright,64] (right linear, no bias)
    const float* we;     // [64] edge weight row
    const float* Wf;     // [64,64]
    const float* bf;     // [64]
    float*       agg;    // [nright,64]
    int          E;
};

__global__ __launch_bounds__(256) void edge_bgc(EdgeArgs g) {
    __shared__ float Ws[64 * 64];
    __shared__ float wes[64];
    __shared__ float bfs[64];
    for (int i = threadIdx.x; i < 4096; i += 256) Ws[i] = g.Wf[i];
    if (threadIdx.x < 64) { wes[threadIdx.x] = g.we[threadIdx.x]; bfs[threadIdx.x] = g.bf[threadIdx.x]; }
    __syncthreads();

    const int wave = threadIdx.x >> 5;
    const int lane = threadIdx.x & 31;
    const int e0   = (blockIdx.x * 8 + wave) * 32;
    if (e0 >= g.E) return;                 // E multiple of 32
    const int hi = lane >> 4;
    const int lo = lane & 15;

    int   liv[2], riv[2];
    float ehv[2];
    const float *Lr[2], *Rr[2];
#pragma unroll
    for (int mt = 0; mt < 2; ++mt) {
        const int e = e0 + mt * 16 + lo;
        liv[mt] = g.li[e];
        riv[mt] = g.ri[e];
        ehv[mt] = g.eh[e];
        Lr[mt] = g.leftL  + (size_t)liv[mt] * 64;
        Rr[mt] = g.rightL + (size_t)riv[mt] * 64;
    }

    v8f acc[2][4] = {{v8f{}, v8f{}, v8f{}, v8f{}},
                     {v8f{}, v8f{}, v8f{}, v8f{}}};
#pragma unroll
    for (int kk = 0; kk < 16; ++kk) {
        const int c0 = 4 * kk + 2 * hi;
        v2f av[2];
#pragma unroll
        for (int mt = 0; mt < 2; ++mt) {
            const float a0 = fmaxf(Lr[mt][c0]     + Rr[mt][c0]     + ehv[mt] * wes[c0],     0.f);
            const float a1 = fmaxf(Lr[mt][c0 + 1] + Rr[mt][c0 + 1] + ehv[mt] * wes[c0 + 1], 0.f);
            av[mt] = v2f{a0, a1};
        }
#pragma unroll
        for (int t = 0; t < 4; ++t) {
            v2f bv = {Ws[c0 * 64 + t * 16 + lo], Ws[(c0 + 1) * 64 + t * 16 + lo]};
            acc[0][t] = __builtin_amdgcn_wmma_f32_16x16x4_f32(
                false, av[0], false, bv, (short)0, acc[0][t], false, false);
            acc[1][t] = __builtin_amdgcn_wmma_f32_16x16x4_f32(
                false, av[1], false, bv, (short)0, acc[1][t], false, false);
        }
    }

#pragma unroll
    for (int mt = 0; mt < 2; ++mt) {
#pragma unroll
        for (int r = 0; r < 8; ++r) {
            const int m   = r + 8 * hi;             // edge slot within tile
            const int rim = __shfl(riv[mt], m, 32); // lane m holds that slot's dest
            float* arow = g.agg + (size_t)rim * 64;
#pragma unroll
            for (int t = 0; t < 4; ++t)
                atomicAdd(&arow[t * 16 + lo], acc[mt][t][r] + bfs[t * 16 + lo]);
        }
    }
}

// ---------------------------------------------------------------------------
// Embedding: x_cat rows 0..NC-1 = cons_x @ Wc + bc (K=1); rows NC.. = var_x @ Wv + bv (K=9)
// ---------------------------------------------------------------------------
__global__ __launch_bounds__(256) void embed_k(const float* cons_x, const float* var_x,
                                               const float* Wc, const float* bc,
                                               const float* Wv, const float* bv, float* xcat) {
    const long i = (long)blockIdx.x * 256 + threadIdx.x;
    if (i >= (long)NN * 64) return;
    const int n = (int)(i >> 6), c = (int)(i & 63);
    float r;
    if (n < NC) {
        r = cons_x[n] * Wc[c] + bc[c];
    } else {
        const float* x = var_x + (size_t)(n - NC) * 9;
        r = bv[c];
#pragma unroll
        for (int k = 0; k < 9; ++k) r += x[k] * Wv[k * 64 + c];
    }
    xcat[i] = r;
}

__global__ __launch_bounds__(256) void edgeh_k(const float* ea, const float* w, const float* b,
                                               float* eh, int E) {
    const int i = blockIdx.x * 256 + threadIdx.x;
    if (i < E) eh[i] = fmaxf(ea[i] * w[0] + b[0], 0.f);
}

__global__ __launch_bounds__(256) void zerof(float* p, long n) {
    const long i = (long)blockIdx.x * 256 + threadIdx.x;
    if (i < n) p[i] = 0.f;
}

// ---------------------------------------------------------------------------
// Attention reductions: sq[0]=||q||^2, sq[1]=||k||^2, svec=sum(k), S=k^T v
// ---------------------------------------------------------------------------
__global__ __launch_bounds__(256) void attn_reduce(const float* q, const float* k, const float* v,
                                                   float* S, float* svec, float* sq, int N) {
    __shared__ float qb[256], kb[256], vb[256];
    float acc[16] = {};
    float lsqq = 0.f, lsqk = 0.f, lsum = 0.f;
    const int tid = threadIdx.x;
    const int d = tid & 63, mg = tid >> 6;
    for (int base = blockIdx.x * 4; base < N; base += gridDim.x * 4) {
        const int r = tid >> 6, c = tid & 63;
        const size_t o = (size_t)(base + r) * 64 + c;
        qb[tid] = q[o]; kb[tid] = k[o]; vb[tid] = v[o];
        __syncthreads();
        lsqq += qb[tid] * qb[tid];
        lsqk += kb[tid] * kb[tid];
        lsum += kb[tid];
#pragma unroll
        for (int j = 0; j < 16; ++j) {
            const int m = mg * 16 + j;
            float a = 0.f;
#pragma unroll
            for (int rr = 0; rr < 4; ++rr) a += kb[rr * 64 + m] * vb[rr * 64 + d];
            acc[j] += a;
        }
        __syncthreads();
    }
#pragma unroll
    for (int j = 0; j < 16; ++j) atomicAdd(&S[(mg * 16 + j) * 64 + d], acc[j]);
    atomicAdd(&svec[d], lsum);
    atomicAdd(&sq[0], lsqq);
    atomicAdd(&sq[1], lsqk);
}

// ---------------------------------------------------------------------------
// Attention epilogue + LN1 + ReLU. Wave per row; hx holds h on entry, x1 on exit.
// ---------------------------------------------------------------------------
__global__ __launch_bounds__(256) void attn_epi(const float* q, const float* QS, const float* v,
                                                const float* svec, const float* sq,
                                                const float* g1, const float* b1,
                                                float* hx, int N) {
    const int lane = threadIdx.x & 31;
    const int wave = threadIdx.x >> 5;
    const float nq = sqrtf(sq[0]) + 1e-8f;
    const float nk = sqrtf(sq[1]) + 1e-8f;
    const float inv = 1.f / (nq * nk);
    const float s0 = svec[lane], s1 = svec[lane + 32];
    const float gg0 = g1[lane], gg1 = g1[lane + 32];
    const float bb0 = b1[lane], bb1 = b1[lane + 32];
    const float fN = (float)N;

    for (int n = blockIdx.x * 8 + wave; n < N; n += gridDim.x * 8) {
        const size_t o = (size_t)n * 64;
        const float q0 = q[o + lane], q1 = q[o + lane + 32];
        float den = q0 * s0 + q1 * s1;
        for (int m = 1; m < 32; m <<= 1) den += __shfl_xor(den, m, 32);
        den = den * inv + fN;
        const float num0 = QS[o + lane] * inv + fN * v[o + lane];
        const float num1 = QS[o + lane + 32] * inv + fN * v[o + lane + 32];
        const float y0 = (num0 / den + hx[o + lane]) * 0.5f;
        const float y1 = (num1 / den + hx[o + lane + 32]) * 0.5f;
        float sum = y0 + y1, ss = y0 * y0 + y1 * y1;
        for (int m = 1; m < 32; m <<= 1) { sum += __shfl_xor(sum, m, 32); ss += __shfl_xor(ss, m, 32); }
        const float mean = sum * (1.f / 64.f);
        const float var  = ss * (1.f / 64.f) - mean * mean;
        const float rstd = rsqrtf(var + LN_EPS);
        hx[o + lane]      = fmaxf((y0 - mean) * rstd * gg0 + bb0, 0.f);
        hx[o + lane + 32] = fmaxf((y1 - mean) * rstd * gg1 + bb1, 0.f);
    }
}

// ---------------------------------------------------------------------------
// Host launch
// ---------------------------------------------------------------------------
static inline void launch_gemm(int epi, bool prerelu, bool concat, const GemmArgs& a,
                               hipStream_t s) {
    const int blocks = (a.rows + 255) / 256;
    if (concat) {
        gemm64<EPI_BIAS_RELU, false, true><<<blocks, 256, 0, s>>>(a);
        return;
    }
    if (prerelu) { gemm64<EPI_BIAS, true, false><<<blocks, 256, 0, s>>>(a); return; }
    switch (epi) {
        case EPI_NONE:      gemm64<EPI_NONE,      false, false><<<blocks, 256, 0, s>>>(a); break;
        case EPI_BIAS:      gemm64<EPI_BIAS,      false, false><<<blocks, 256, 0, s>>>(a); break;
        case EPI_BIAS_RELU: gemm64<EPI_BIAS_RELU, false, false><<<blocks, 256, 0, s>>>(a); break;
        case EPI_LN_RELU:   gemm64<EPI_LN_RELU,   false, false><<<blocks, 256, 0, s>>>(a); break;
        case EPI_MIX:       gemm64<EPI_MIX,       false, false><<<blocks, 256, 0, s>>>(a); break;
    }
}

extern "C" void kernel_launch(void* const* d_in, const int* in_sizes, int n_in,
                              void* d_out, int out_size, void* d_ws, size_t ws_size,
                              hipStream_t stream) {
    (void)in_sizes; (void)n_in; (void)out_size; (void)ws_size;
    #define F(i) ((const float*)d_in[i])
    const float* cons_x    = F(0);
    const int*   ei        = (const int*)d_in[1];
    const int*   ei0       = ei;          // cons index per edge
    const int*   ei1       = ei + NE;     // var index per edge
    const float* edge_attr = F(2);
    const float* var_x     = F(3);
    // params (flat leaf order of setup_inputs)
    const float *veW=F(4), *veB=F(5), *ceW=F(6), *ceB=F(7);
    const float *fc0W=F(8), *fc0B=F(9), *ln0g=F(10), *ln0b=F(11);
    const float *WqW=F(12), *WqB=F(13), *WkW=F(14), *WkB=F(15), *WvW=F(16), *WvB=F(17);
    const float *ln1g=F(18), *ln1b=F(19);
    const float *c1W=F(20), *c1B=F(21), *c2W=F(22), *c2B=F(23);
    const float *geW=F(24), *geB=F(25);
    const float *v1W=F(26), *v1B=F(27), *v2W=F(28), *v2B=F(29);
    // v2c
    const float *aLW=F(30), *aLB=F(31), *aEW=F(32), *aRW=F(33);
    const float *aFW=F(34), *aFB=F(35), *aPW=F(36), *aPB=F(37);
    const float *aO1W=F(38), *aO1B=F(39), *aO2W=F(40), *aO2B=F(41);
    // c2v
    const float *bLW=F(42), *bLB=F(43), *bEW=F(44), *bRW=F(45);
    const float *bFW=F(46), *bFB=F(47), *bPW=F(48), *bPB=F(49);
    const float *bO1W=F(50), *bO1B=F(51), *bO2W=F(52), *bO2B=F(53);
    const float *ovW=F(54), *ovB=F(55), *ocW=F(56), *ocB=F(57);
    #undef F

    float* ws = (float*)d_ws;
    const size_t SZ = (size_t)NN * H;
    float* B0 = ws;            // x_cat (cons_e rows 0..NC, var_e rows NC..)
    float* B1 = B0 + SZ;       // h, later x1
    float* B2 = B1 + SZ;       // q, later tmp / t1
    float* B3 = B2 + SZ;       // k, later left_lin
    float* B4 = B3 + SZ;       // v, later right_lin
    float* B5 = B4 + SZ;       // QS, later agg / t2
    float* B6 = B5 + SZ;       // cons_h [NC,64]
    float* B7 = B6 + (size_t)NC * H;   // var_h [NV,64]
    float* B8 = B7 + (size_t)NV * H;   // edge_h [NE]
    float* Smat = B8 + NE;     // [64,64]
    float* svec = Smat + 4096; // [64]
    float* sq   = svec + 64;   // [2]
    float* dout = (float*)d_out;

    // 0) zero attention scratch
    zerof<<<(4162 + 255) / 256, 256, 0, stream>>>(Smat, 4162);
    // 1) embeddings
    embed_k<<<(int)(((long)NN * 64 + 255) / 256), 256, 0, stream>>>(
        cons_x, var_x, ceW, ceB, veW, veB, B0);
    // 2) h = relu(LN0(x_cat @ fc0 + b))
    { GemmArgs a{B0,nullptr,fc0W,fc0B,B1,ln0g,ln0b,nullptr,nullptr,NN};
      launch_gemm(EPI_LN_RELU,false,false,a,stream); }
    // 3) q,k,v
    { GemmArgs a{B1,nullptr,WqW,WqB,B2,nullptr,nullptr,nullptr,nullptr,NN};
      launch_gemm(EPI_BIAS,false,false,a,stream); }
    { GemmArgs a{B1,nullptr,WkW,WkB,B3,nullptr,nullptr,nullptr,nullptr,NN};
      launch_gemm(EPI_BIAS,false,false,a,stream); }
    { GemmArgs a{B1,nullptr,WvW,WvB,B4,nullptr,nullptr,nullptr,nullptr,NN};
      launch_gemm(EPI_BIAS,false,false,a,stream); }
    // 4) reductions: ||q||^2, ||k||^2, sum(k), k^T v
    attn_reduce<<<1024, 256, 0, stream>>>(B2, B3, B4, Smat, svec, sq, NN);
    // 5) QS = q @ S (raw; scaling applied in epilogue)
    { GemmArgs a{B2,nullptr,Smat,nullptr,B5,nullptr,nullptr,nullptr,nullptr,NN};
      launch_gemm(EPI_NONE,false,false,a,stream); }
    // 6) x1 = relu(LN1((num/den + h)/2))  (in-place into B1)
    attn_epi<<<2048, 256, 0, stream>>>(B2, B5, B4, svec, sq, ln1g, ln1b, B1, NN);

    // 7) GNN feature stacks
    { GemmArgs a{B0,nullptr,c1W,c1B,B2,nullptr,nullptr,nullptr,nullptr,NC};
      launch_gemm(EPI_BIAS_RELU,false,false,a,stream); }
    { GemmArgs a{B2,nullptr,c2W,c2B,B6,nullptr,nullptr,nullptr,nullptr,NC};
      launch_gemm(EPI_BIAS_RELU,false,false,a,stream); }
    { GemmArgs a{B0 + (size_t)NC*H,nullptr,v1W,v1B,B2,nullptr,nullptr,nullptr,nullptr,NV};
      launch_gemm(EPI_BIAS_RELU,false,false,a,stream); }
    { GemmArgs a{B2,nullptr,v2W,v2B,B7,nullptr,nullptr,nullptr,nullptr,NV};
      launch_gemm(EPI_BIAS_RELU,false,false,a,stream); }
    edgeh_k<<<(NE + 255) / 256, 256, 0, stream>>>(edge_attr, geW, geB, B8, NE);

    const int edgeBlocks = (NE / 32 + 7) / 8;

    // 8) v2c: left=var_h (li=ei1), right=cons_h (ri=ei0), agg over cons
    { GemmArgs a{B7,nullptr,aLW,aLB,B3,nullptr,nullptr,nullptr,nullptr,NV};
      launch_gemm(EPI_BIAS,false,false,a,stream); }
    { GemmArgs a{B6,nullptr,aRW,nullptr,B4,nullptr,nullptr,nullptr,nullptr,NC};
      launch_gemm(EPI_NONE,false,false,a,stream); }
    zerof<<<(int)(((long)NC * H + 255) / 256), 256, 0, stream>>>(B5, (long)NC * H);
    { EdgeArgs e{ei1, ei0, B8, B3, B4, aEW, aFW, aFB, B5, NE};
      edge_bgc<<<edgeBlocks, 256, 0, stream>>>(e); }
    { GemmArgs a{B5,nullptr,aPW,aPB,B2,nullptr,nullptr,nullptr,nullptr,NC};   // t1 = relu(agg)@post
      launch_gemm(EPI_BIAS,true,false,a,stream); }
    { GemmArgs a{B2,B6,aO1W,aO1B,B5,nullptr,nullptr,nullptr,nullptr,NC};      // t2 = relu([t1,cons_h]@out1)
      launch_gemm(EPI_BIAS_RELU,false,true,a,stream); }
    { GemmArgs a{B5,nullptr,aO2W,aO2B,B6,nullptr,nullptr,nullptr,nullptr,NC}; // cons_h = t2@out2
      launch_gemm(EPI_BIAS,false,false,a,stream); }

    // 9) c2v: left=cons_h (li=ei0), right=var_h (ri=ei1), agg over vars
    { GemmArgs a{B6,nullptr,bLW,bLB,B3,nullptr,nullptr,nullptr,nullptr,NC};
      launch_gemm(EPI_BIAS,false,false,a,stream); }
    { GemmArgs a{B7,nullptr,bRW,nullptr,B4,nullptr,nullptr,nullptr,nullptr,NV};
      launch_gemm(EPI_NONE,false,false,a,stream); }
    zerof<<<(int)(((long)NV * H + 255) / 256), 256, 0, stream>>>(B5, (long)NV * H);
    { EdgeArgs e{ei0, ei1, B8, B3, B4, bEW, bFW, bFB, B5, NE};
      edge_bgc<<<edgeBlocks, 256, 0, stream>>>(e); }
    { GemmArgs a{B5,nullptr,bPW,bPB,B2,nullptr,nullptr,nullptr,nullptr,NV};
      launch_gemm(EPI_BIAS,true,false,a,stream); }
    { GemmArgs a{B2,B7,bO1W,bO1B,B5,nullptr,nullptr,nullptr,nullptr,NV};
      launch_gemm(EPI_BIAS_RELU,false,true,a,stream); }
    { GemmArgs a{B5,nullptr,bO2W,bO2B,B7,nullptr,nullptr,nullptr,nullptr,NV}; // var_h = t2@out2
      launch_gemm(EPI_BIAS,false,false,a,stream); }

    // 10) outputs: x = 0.8*relu(h@Wo+b) + 0.2*x1 ; dup vars/cons sections
    { GemmArgs a{B6,nullptr,ocW,ocB,dout,nullptr,nullptr,B1,
                 dout + (size_t)(NN + NV) * H, NC};
      launch_gemm(EPI_MIX,false,false,a,stream); }
    { GemmArgs a{B7,nullptr,ovW,ovB,dout + (size_t)NC * H,nullptr,nullptr,
                 B1 + (size_t)NC * H, dout + (size_t)NN * H, NV};
      launch_gemm(EPI_MIX,false,false,a,stream); }
}